// Decoder_82635170775618
// MI455X (gfx1250) — compile-verified
//
#include <hip/hip_runtime.h>
#include <hip/hip_bf16.h>
#include <math.h>

#define B    32
#define M    2048
#define KTRIP 4
#define E    128
#define G    128
#define V    50000
#define HOPS 3
#define MCHUNKS 16            // m-dimension split for hop kernels (2048/128)

typedef float v2f __attribute__((ext_vector_type(2)));
typedef float v8f __attribute__((ext_vector_type(8)));

// ---------------------------------------------------------------- GRU cell
__global__ void __launch_bounds__(G)
gru_kernel(const int* __restrict__ y_prev, const float* __restrict__ h_prev,
           const float* __restrict__ emb0, const float* __restrict__ W_ih,
           const float* __restrict__ W_hh, const float* __restrict__ b_ih,
           const float* __restrict__ b_hh, float* __restrict__ out_h,
           float* __restrict__ q, float* __restrict__ cat)
{
    const int b = blockIdx.x;
    const int g = threadIdx.x;
    __shared__ float xs[E];
    __shared__ float hs[G];
    const int y = y_prev[b];
    xs[g] = emb0[(size_t)y * E + g];
    hs[g] = h_prev[b * G + g];
    __syncthreads();
    float ir = 0.f, iz = 0.f, inn = 0.f, hr = 0.f, hz = 0.f, hn = 0.f;
    for (int e = 0; e < E; ++e) {
        const float xe = xs[e], he = hs[e];
        ir  += xe * W_ih[(0 * G + g) * E + e];
        iz  += xe * W_ih[(1 * G + g) * E + e];
        inn += xe * W_ih[(2 * G + g) * E + e];
        hr  += he * W_hh[(0 * G + g) * G + e];
        hz  += he * W_hh[(1 * G + g) * G + e];
        hn  += he * W_hh[(2 * G + g) * G + e];
    }
    ir += b_ih[g]; iz += b_ih[G + g]; inn += b_ih[2 * G + g];
    hr += b_hh[g]; hz += b_hh[G + g]; hn  += b_hh[2 * G + g];
    const float r = 1.f / (1.f + expf(-(ir + hr)));
    const float z = 1.f / (1.f + expf(-(iz + hz)));
    const float n = tanhf(inn + r * hn);
    const float hnew = (1.f - z) * n + z * hs[g];
    out_h[b * G + g]   = hnew;
    q[b * G + g]       = hnew;
    cat[b * 2 * E + g] = hnew;   // cat[:, 0:128] = h
}

// ------------------------------------------------- hop: attention scores
// grid = B * MCHUNKS, block = 256 (8 waves); one wave per memory slot m.
__global__ void __launch_bounds__(256)
hop_scores(const int* __restrict__ ctx, const float* __restrict__ embA,
           const float* __restrict__ q, float* __restrict__ p)
{
    const int b     = blockIdx.x >> 4;
    const int mbase = (blockIdx.x & (MCHUNKS - 1)) * 128;
    __shared__ __align__(16) float qs[E];
    const int tid = threadIdx.x;
    if (tid < E) qs[tid] = q[b * E + tid];
    __syncthreads();
    const int wave = tid >> 5, lane = tid & 31;
    const int e0 = lane * 4;
    const float4 qv = *reinterpret_cast<const float4*>(qs + e0);
    for (int i = 0; i < 16; ++i) {
        const int m = mbase + wave * 16 + i;
        const int* cp = ctx + ((size_t)b * M + m) * KTRIP;
        const int i0 = cp[0], i1 = cp[1], i2 = cp[2], i3 = cp[3];
        const float4 v0 = *reinterpret_cast<const float4*>(embA + (size_t)i0 * E + e0);
        const float4 v1 = *reinterpret_cast<const float4*>(embA + (size_t)i1 * E + e0);
        const float4 v2 = *reinterpret_cast<const float4*>(embA + (size_t)i2 * E + e0);
        const float4 v3 = *reinterpret_cast<const float4*>(embA + (size_t)i3 * E + e0);
        const float sx = v0.x + v1.x + v2.x + v3.x;
        const float sy = v0.y + v1.y + v2.y + v3.y;
        const float sz = v0.z + v1.z + v2.z + v3.z;
        const float sw = v0.w + v1.w + v2.w + v3.w;
        float s = sx * qv.x + sy * qv.y + sz * qv.z + sw * qv.w;
        #pragma unroll
        for (int off = 16; off > 0; off >>= 1) s += __shfl_xor(s, off, 32);
        if (lane == 0) p[(size_t)b * M + m] = s;
    }
}

// ------------------------------------------- generic row softmax, in place
// blockDim.x chosen per launch (256 for M=2048 rows, 1024 for V=50000 rows).
__global__ void
softmax_rows(float* __restrict__ p, float* __restrict__ out_copy, int rowlen)
{
    const int b = blockIdx.x, tid = threadIdx.x, nt = blockDim.x;
    __shared__ float red[1024];
    float* row = p + (size_t)b * rowlen;
    float mx = -INFINITY;
    for (int i = tid; i < rowlen; i += nt) mx = fmaxf(mx, row[i]);
    red[tid] = mx; __syncthreads();
    for (int s = nt >> 1; s > 0; s >>= 1) {
        if (tid < s) red[tid] = fmaxf(red[tid], red[tid + s]);
        __syncthreads();
    }
    const float rmax = red[0];
    __syncthreads();
    float sum = 0.f;
    for (int i = tid; i < rowlen; i += nt) sum += expf(row[i] - rmax);
    red[tid] = sum; __syncthreads();
    for (int s = nt >> 1; s > 0; s >>= 1) {
        if (tid < s) red[tid] += red[tid + s];
        __syncthreads();
    }
    const float inv = 1.f / red[0];
    __syncthreads();
    for (int i = tid; i < rowlen; i += nt) {
        const float a = expf(row[i] - rmax) * inv;
        row[i] = a;
        if (out_copy) out_copy[(size_t)b * rowlen + i] = a;
    }
}

// ---------------------- hop: weighted memory read (partial over m chunk)
// grid = B * MCHUNKS, block = 128; each block reduces 128 memory slots.
__global__ void __launch_bounds__(128)
hop_read_partial(const int* __restrict__ ctx, const float* __restrict__ embC,
                 const float* __restrict__ attn, float* __restrict__ opart)
{
    const int b = blockIdx.x >> 4;
    const int c = blockIdx.x & (MCHUNKS - 1);
    const int e = threadIdx.x;
    const int mbase = c * 128;
    __shared__ float a_s[128];
    __shared__ int   id_s[128 * KTRIP];
    a_s[e] = attn[(size_t)b * M + mbase + e];
    const int* cp = ctx + ((size_t)b * M + mbase) * KTRIP;
    for (int j = e; j < 128 * KTRIP; j += 128) id_s[j] = cp[j];
    __syncthreads();
    float acc = 0.f;
    #pragma unroll 4
    for (int i = 0; i < 128; ++i) {
        const float w = a_s[i];
        const int* id = id_s + i * KTRIP;
        acc += w * (embC[(size_t)id[0] * E + e] + embC[(size_t)id[1] * E + e]
                  + embC[(size_t)id[2] * E + e] + embC[(size_t)id[3] * E + e]);
    }
    opart[((size_t)b * MCHUNKS + c) * E + e] = acc;
}

// ------------------------------- hop: combine partials, update q (and o1)
__global__ void __launch_bounds__(E)
hop_combine(const float* __restrict__ opart, float* __restrict__ q,
            float* __restrict__ cat, int hop)
{
    const int b = blockIdx.x, e = threadIdx.x;
    float o = 0.f;
    #pragma unroll
    for (int c = 0; c < MCHUNKS; ++c) o += opart[((size_t)b * MCHUNKS + c) * E + e];
    q[b * E + e] += o;
    if (hop == 0) cat[b * 2 * E + E + e] = o;   // cat[:, 128:256] = o1
}

// ----------------------------- vocab logits via V_WMMA_F32_16X16X4_F32
// C tile 16x16 per wave; block = 8 waves -> 32 rows x 64 vocab cols.
// A (16x4 f32): lanes 0-15 M=lane K={0,1}; lanes 16-31 M=lane-16 K={2,3}.
// B (4x16 f32): lanes 0-15 N=lane K={0,1}; lanes 16-31 N=lane-16 K={2,3}.
// C (16x16 f32): VGPR r -> lanes 0-15 (M=r,N=lane), lanes 16-31 (M=r+8).
__global__ void __launch_bounds__(256)
vocab_logits(const float* __restrict__ cat, const float* __restrict__ Wv,
             const float* __restrict__ bv, float* __restrict__ logits)
{
    __shared__ float cs[B * 2 * E];              // 32 x 256 = 32 KB
    const int tid = threadIdx.x;
    for (int i = tid; i < B * 2 * E; i += 256) cs[i] = cat[i];
    __syncthreads();
    const int wave = tid >> 5, lane = tid & 31;
    const int mt = wave & 1;                     // row tile 0/1
    const int nt = wave >> 1;                    // col tile 0..3
    const int n0 = blockIdx.x * 64 + nt * 16;
    if (n0 >= V) return;                         // uniform per wave, V % 16 == 0
    const int half = lane >> 4;
    const int l15  = lane & 15;
    const float* arow = cs + (mt * 16 + l15) * 256 + half * 2;
    const int nn = n0 + l15;
    const float* brow = Wv + (size_t)nn * 256 + half * 2;
    v8f c = {0.f, 0.f, 0.f, 0.f, 0.f, 0.f, 0.f, 0.f};
    #pragma unroll 8
    for (int k = 0; k < 256; k += 4) {
        v2f a, bb;
        a.x  = arow[k]; a.y  = arow[k + 1];
        bb.x = brow[k]; bb.y = brow[k + 1];
        c = __builtin_amdgcn_wmma_f32_16x16x4_f32(false, a, false, bb,
                                                  (short)0, c, false, false);
    }
    const float bvn = bv[nn];
    const int mbase = mt * 16 + half * 8;
    #pragma unroll
    for (int r = 0; r < 8; ++r)
        logits[(size_t)(mbase + r) * V + nn] = c[r] + bvn;
}

// --------------------------------------------------------------- launcher
extern "C" void kernel_launch(void* const* d_in, const int* in_sizes, int n_in,
                              void* d_out, int out_size, void* d_ws, size_t ws_size,
                              hipStream_t stream)
{
    const int*   ctx    = (const int*)  d_in[0];
    const float* h_prev = (const float*)d_in[1];
    const int*   y_prev = (const int*)  d_in[2];
    const float* emb    = (const float*)d_in[3];
    const float* W_ih   = (const float*)d_in[4];
    const float* W_hh   = (const float*)d_in[5];
    const float* b_ih   = (const float*)d_in[6];
    const float* b_hh   = (const float*)d_in[7];
    const float* W_voc  = (const float*)d_in[8];
    const float* b_voc  = (const float*)d_in[9];

    float* out      = (float*)d_out;
    float* out_h    = out;                               // [B, G]
    float* out_pv   = out + B * G;                       // [B, V]
    float* out_attn = out + B * G + (size_t)B * V;       // [B, M]

    float* wsf   = (float*)d_ws;                         // ~570 KB used
    float* q     = wsf;                                  // [B, E]
    float* cat   = q + B * E;                            // [B, 2E]
    float* p     = cat + B * 2 * E;                      // [B, M]
    float* opart = p + (size_t)B * M;                    // [B, MCHUNKS, E]

    gru_kernel<<<B, G, 0, stream>>>(y_prev, h_prev, emb, W_ih, W_hh,
                                    b_ih, b_hh, out_h, q, cat);

    for (int hop = 0; hop < HOPS; ++hop) {
        const float* embA = emb + (size_t)hop * V * E;
        const float* embC = emb + (size_t)(hop + 1) * V * E;
        hop_scores<<<B * MCHUNKS, 256, 0, stream>>>(ctx, embA, q, p);
        softmax_rows<<<B, 256, 0, stream>>>(
            p, (hop == HOPS - 1) ? out_attn : (float*)nullptr, M);
        hop_read_partial<<<B * MCHUNKS, 128, 0, stream>>>(ctx, embC, p, opart);
        hop_combine<<<B, E, 0, stream>>>(opart, q, cat, hop);
    }

    vocab_logits<<<(V + 63) / 64, 256, 0, stream>>>(cat, W_voc, b_voc, out_pv);
    softmax_rows<<<B, 1024, 0, stream>>>(out_pv, (float*)nullptr, V);
}